// O3CGMLP_7035156431111
// MI455X (gfx1250) — compile-verified
//
#include <hip/hip_runtime.h>
#include <math.h>

// ---------------------------------------------------------------------------
// O3 Clifford-MLP (B=32768, 3 -> 32 -> 1 channels) on MI455X / gfx1250.
// Both geometric-product contractions are folded into f32 GEMMs and executed
// with V_WMMA_F32_16X16X4_F32 (native fp32 WMMA; keeps reference precision).
// ---------------------------------------------------------------------------

typedef float v2f __attribute__((ext_vector_type(2)));
typedef float v8f __attribute__((ext_vector_type(8)));

#define B_PTS   32768
#define FIN1    3
#define FH      32
#define FOUT    1
#define N1      256      // FH * 8 output columns of layer-1 GEMM
#define KGEO1   192      // FIN1 * 64
#define KLEND1  216      // KGEO1 + FIN1*8   (linear-term region end; k==216 is bias)
#define KP1     224      // padded K (multiple of 4)
#define NSTEP1  16       // N1 / 16
#define KSTEP1  56       // KP1 / 4

#define KGEO2   2048     // FH * 64
#define KLEND2  2304     // KGEO2 + FH*8     (k==2304 is bias)
#define KP2     2308
#define KSTEP2  577      // KP2 / 4

#define QW1     112      // KP1 / 2 paired rows
#define QW2     1154     // KP2 / 2

#define XSTR    260      // padded LDS row stride (4 mod 64 -> conflict-free gathers)

#define INV_SQRT2 0.70710678118654752f
#define EPS 1e-6f

// grade of blade i: [0,1,1,1,2,2,2,3] packed in nibbles
__device__ __forceinline__ int gof(int i) { return (int)((0x32221110u >> (i * 4)) & 0xFu); }
__device__ __forceinline__ float sigm(float x) { return 1.0f / (1.0f + __expf(-x)); }

// D = A(16x4 f32) * B(4x16 f32) + C(16x16 f32), full-precision CDNA5 WMMA.
// A layout (ISA 7.12.2): lane m (m+16) row M=m, VGPR0=K{0|2}, VGPR1=K{1|3}.
// B layout assumed symmetric: VGPR0 rows K={0|2}, VGPR1 rows K={1|3}, N=lane&15.
__device__ __forceinline__ v8f wmma4(v2f a, v2f b, v8f c) {
#if defined(__HIP_DEVICE_COMPILE__) && defined(__gfx1250__)
  return __builtin_amdgcn_wmma_f32_16x16x4_f32(false, a, false, b, (short)0, c, false, false);
#else
  (void)a; (void)b;
  return c;  // host-pass stub
#endif
}

// ---------------------------------------------------------------------------
// Folded GEMM weight:  col = o*8 + j
//   k <  KGEO        : cayley[i,j,kk] * w_geo[o,n,g(i),g(j),g(kk)],  k=n*64+i*8+kk
//   k in lin region  : (j'==j) ? wl[o,n,g(j)] : 0,                   k-KGEO=n*8+j'
//   k == bias index  : (j==0) ? bl[o] : 0
//   else             : 0 (padding)
// ---------------------------------------------------------------------------
__device__ __forceinline__ float wval(int k, int col, int fin, int fout, int kgeo,
                                      const float* cay, const float* wgeo,
                                      const float* wl, const float* bl) {
  int o = col >> 3, j = col & 7;
  if (o >= fout) return 0.0f;
  if (k < kgeo) {
    int n = k >> 6, i = (k >> 3) & 7, kk = k & 7;
    float c = cay[(i * 8 + j) * 8 + kk];
    return c * wgeo[(o * fin + n) * 64 + gof(i) * 16 + gof(j) * 4 + gof(kk)];
  }
  int t = k - kgeo;
  if (t < fin * 8) {
    int n = t >> 3, jp = t & 7;
    return (jp == j) ? wl[(o * fin + n) * 4 + gof(j)] : 0.0f;
  }
  if (t == fin * 8) return (j == 0) ? bl[o] : 0.0f;
  return 0.0f;
}

__global__ __launch_bounds__(256) void build_weights(
    const float* __restrict__ cay,
    const float* __restrict__ wgeo1, const float* __restrict__ wl1, const float* __restrict__ bl1,
    const float* __restrict__ wgeo2, const float* __restrict__ wl2, const float* __restrict__ bl2,
    float* __restrict__ wp1, float* __restrict__ wp2) {
  int t = blockIdx.x * 256 + threadIdx.x;
  if (t < QW1 * N1) {                       // layer-1 weights, paired-row layout
    int q = t >> 8, col = t & 255;
    wp1[2 * t + 0] = wval(2 * q,     col, FIN1, FH, KGEO1, cay, wgeo1, wl1, bl1);
    wp1[2 * t + 1] = wval(2 * q + 1, col, FIN1, FH, KGEO1, cay, wgeo1, wl1, bl1);
  } else {
    int t2 = t - QW1 * N1;
    if (t2 < QW2 * 16) {                    // layer-2 weights (N padded 8 -> 16)
      int q = t2 >> 4, col = t2 & 15;
      wp2[2 * t2 + 0] = wval(2 * q,     col, FH, FOUT, KGEO2, cay, wgeo2, wl2, bl2);
      wp2[2 * t2 + 1] = wval(2 * q + 1, col, FH, FOUT, KGEO2, cay, wgeo2, wl2, bl2);
    }
  }
}

// ---------------------------------------------------------------------------
// Layer 1: one wave per 16 points. Prep x0 = embed(points@P) and
// xr = normalize(mv_linear(x0, wr1)) into LDS, then 56 K-steps x 16 N-tiles
// of f32 WMMA with on-the-fly A construction.
// ---------------------------------------------------------------------------
__global__ __launch_bounds__(32) void gemm_fcsgp1(
    const float* __restrict__ points, const float* __restrict__ P,
    const float* __restrict__ wr1, const float* __restrict__ an1,
    const float* __restrict__ beta, const float* __restrict__ wp1,
    float* __restrict__ x_mid) {
  __shared__ float xA[16 * 25];
  __shared__ float xR[16 * 25];
  __shared__ float stage[16 * 256];

  const int lane = threadIdx.x;
  const int rowbase = blockIdx.x * 16;
  const int c0 = lane & 15;
  const int half = lane >> 4;

  if (lane < 16) {
    const int m = lane;
    const int row = rowbase + m;
    float x0[FIN1][8];
    for (int n = 0; n < FIN1; ++n) {
      float p0 = points[(row * 3 + n) * 3 + 0];
      float p1 = points[(row * 3 + n) * 3 + 1];
      float p2 = points[(row * 3 + n) * 3 + 2];
      x0[n][0] = 0.0f;
      for (int c = 0; c < 3; ++c)
        x0[n][1 + c] = p0 * P[0 * 3 + c] + p1 * P[1 * 3 + c] + p2 * P[2 * 3 + c];
      x0[n][4] = 0.0f; x0[n][5] = 0.0f; x0[n][6] = 0.0f; x0[n][7] = 0.0f;
      for (int i = 0; i < 8; ++i) xA[m * 25 + n * 8 + i] = x0[n][i];
    }
    for (int mo = 0; mo < FIN1; ++mo) {
      float r[8];
      for (int i = 0; i < 8; ++i) {
        float s = 0.0f;
        for (int n = 0; n < FIN1; ++n) s += x0[n][i] * wr1[(mo * FIN1 + n) * 4 + gof(i)];
        r[i] = s;
      }
      float z[8];
      for (int jj = 0; jj < 8; ++jj) {
        float s = 0.0f;
        for (int i = 0; i < 8; ++i) s += r[i] * beta[i * 8 + jj];
        z[jj] = s;
      }
      float q[4];
      q[0] = r[0] * z[0];
      q[1] = r[1] * z[1] + r[2] * z[2] + r[3] * z[3];
      q[2] = r[4] * z[4] + r[5] * z[5] + r[6] * z[6];
      q[3] = r[7] * z[7];
      float den[4];
      for (int g = 0; g < 4; ++g) {
        float nr = sqrtf(fabsf(q[g]));
        den[g] = sigm(an1[mo * 4 + g]) * (nr - 1.0f) + 1.0f + EPS;
      }
      for (int i = 0; i < 8; ++i) xR[m * 25 + mo * 8 + i] = r[i] / den[gof(i)];
    }
  }
  __syncthreads();

  v8f acc[NSTEP1];
  const v8f vzero = {0, 0, 0, 0, 0, 0, 0, 0};
#pragma unroll
  for (int nt = 0; nt < NSTEP1; ++nt) acc[nt] = vzero;

#pragma unroll 1
  for (int kb = 0; kb < KSTEP1; ++kb) {
    const int kbase = kb * 4;            // wave-uniform region select
    const int k0 = kbase + 2 * half;     // this lane's K pair (even)
    v2f a = {0.0f, 0.0f};
    if (kbase < KGEO1) {
      int n = k0 >> 6, i = (k0 >> 3) & 7, kk = k0 & 7;
      float xv = xA[c0 * 25 + n * 8 + i];
      a.x = xv * xR[c0 * 25 + n * 8 + kk];
      a.y = xv * xR[c0 * 25 + n * 8 + kk + 1];
    } else if (kbase < KLEND1) {
      int t = k0 - KGEO1; int n = t >> 3, jp = t & 7;
      a.x = xA[c0 * 25 + n * 8 + jp];
      a.y = xA[c0 * 25 + n * 8 + jp + 1];
    } else if (kbase == KLEND1) {
      a.x = (half == 0) ? 1.0f : 0.0f;   // bias column; rest is zero padding
    }
    const int qrow = kb * 2 + half;
#pragma unroll
    for (int nt = 0; nt < NSTEP1; ++nt) {
      v2f b = *reinterpret_cast<const v2f*>(wp1 + 2 * ((size_t)qrow * N1 + nt * 16 + c0));
      acc[nt] = wmma4(a, b, acc[nt]);
    }
  }

  // C/D layout: VGPR r, lanes 0-15 -> M=r, lanes 16-31 -> M=r+8, N = lane&15.
#pragma unroll
  for (int nt = 0; nt < NSTEP1; ++nt) {
#pragma unroll
    for (int r = 0; r < 8; ++r)
      stage[(r + 8 * half) * 256 + nt * 16 + c0] = acc[nt][r] * INV_SQRT2;
  }
  __syncthreads();

  const float4* s4 = reinterpret_cast<const float4*>(stage);
  float4* d4 = reinterpret_cast<float4*>(x_mid + (size_t)rowbase * 256);
  for (int t = lane; t < 1024; t += 32) d4[t] = s4[t];
}

// ---------------------------------------------------------------------------
// 5 fused MV-SiLU layers, one thread per (point, channel).
// ---------------------------------------------------------------------------
__global__ __launch_bounds__(256) void silu5(
    float* __restrict__ x_mid, const float* __restrict__ beta,
    const float* __restrict__ sa, const float* __restrict__ sb) {
  __shared__ float bs[64];
  __shared__ float as_[5 * FH * 4];
  __shared__ float bbs[5 * FH * 4];
  const int tid = threadIdx.x;
  if (tid < 64) bs[tid] = beta[tid];
  for (int t = tid; t < 5 * FH * 4; t += 256) { as_[t] = sa[t]; bbs[t] = sb[t]; }
  __syncthreads();

  const int gid = blockIdx.x * 256 + tid;
  const int b = gid >> 5, o = gid & 31;
  float* px = x_mid + (size_t)b * 256 + o * 8;
  float x[8];
  {
    float4 lo = reinterpret_cast<const float4*>(px)[0];
    float4 hi = reinterpret_cast<const float4*>(px)[1];
    x[0] = lo.x; x[1] = lo.y; x[2] = lo.z; x[3] = lo.w;
    x[4] = hi.x; x[5] = hi.y; x[6] = hi.z; x[7] = hi.w;
  }
#pragma unroll
  for (int l = 0; l < 5; ++l) {
    float z[8];
#pragma unroll
    for (int jj = 0; jj < 8; ++jj) {
      float s = 0.0f;
#pragma unroll
      for (int i = 0; i < 8; ++i) s += x[i] * bs[i * 8 + jj];
      z[jj] = s;
    }
    float inv[4];
    inv[0] = x[0];
    inv[1] = x[1] * z[1] + x[2] * z[2] + x[3] * z[3];
    inv[2] = x[4] * z[4] + x[5] * z[5] + x[6] * z[6];
    inv[3] = x[7] * z[7];
    float gate[4];
#pragma unroll
    for (int g = 0; g < 4; ++g)
      gate[g] = sigm(as_[(l * FH + o) * 4 + g] * inv[g] + bbs[(l * FH + o) * 4 + g]);
    x[0] *= gate[0];
    x[1] *= gate[1]; x[2] *= gate[1]; x[3] *= gate[1];
    x[4] *= gate[2]; x[5] *= gate[2]; x[6] *= gate[2];
    x[7] *= gate[3];
  }
  float4 lo = make_float4(x[0], x[1], x[2], x[3]);
  float4 hi = make_float4(x[4], x[5], x[6], x[7]);
  reinterpret_cast<float4*>(px)[0] = lo;
  reinterpret_cast<float4*>(px)[1] = hi;
}

// ---------------------------------------------------------------------------
// Layer 2: one wave per 16 points, K=2308, single N-tile; only column j=7
// of the 16x16 result feeds the loss, emitted directly into d_out[1..].
// ---------------------------------------------------------------------------
__global__ __launch_bounds__(32) void gemm_fcsgp2(
    const float* __restrict__ x_mid, const float* __restrict__ wr2,
    const float* __restrict__ an2, const float* __restrict__ beta,
    const float* __restrict__ wp2, const float* __restrict__ products,
    const float* __restrict__ detP, float* __restrict__ out) {
  __shared__ float xIn[16 * XSTR];
  __shared__ float wr2s[FH * FH * 4];
  __shared__ float xRs[16 * XSTR];
  __shared__ float betas[64];

  const int lane = threadIdx.x;
  const int rowbase = blockIdx.x * 16;
  const int c0 = lane & 15;
  const int half = lane >> 4;

  {
    const float4* src = reinterpret_cast<const float4*>(x_mid + (size_t)rowbase * 256);
    for (int t = lane; t < 16 * 64; t += 32) {
      int r = t >> 6, c4 = t & 63;
      reinterpret_cast<float4*>(xIn + r * XSTR)[c4] = src[r * 64 + c4];
    }
    const float4* wsrc = reinterpret_cast<const float4*>(wr2);
    float4* wdst = reinterpret_cast<float4*>(wr2s);
    for (int t = lane; t < (FH * FH * 4) / 4; t += 32) wdst[t] = wsrc[t];
    if (lane < 16)
      reinterpret_cast<float4*>(betas)[lane] = reinterpret_cast<const float4*>(beta)[lane];
  }
  __syncthreads();

  {  // xr2 = normalize(mv_linear(xIn, wr2)); lane halves split the 32 channels
    const int m = c0;
    for (int mo = half * 16; mo < half * 16 + 16; ++mo) {
      float r[8];
      for (int i = 0; i < 8; ++i) {
        float s = 0.0f;
        const int g = gof(i);
        for (int n = 0; n < FH; ++n)
          s += xIn[m * XSTR + n * 8 + i] * wr2s[(mo * FH + n) * 4 + g];
        r[i] = s;
      }
      float z[8];
      for (int jj = 0; jj < 8; ++jj) {
        float s = 0.0f;
        for (int i = 0; i < 8; ++i) s += r[i] * betas[i * 8 + jj];
        z[jj] = s;
      }
      float q[4];
      q[0] = r[0] * z[0];
      q[1] = r[1] * z[1] + r[2] * z[2] + r[3] * z[3];
      q[2] = r[4] * z[4] + r[5] * z[5] + r[6] * z[6];
      q[3] = r[7] * z[7];
      float den[4];
      for (int g = 0; g < 4; ++g) {
        float nr = sqrtf(fabsf(q[g]));
        den[g] = sigm(an2[mo * 4 + g]) * (nr - 1.0f) + 1.0f + EPS;
      }
      for (int i = 0; i < 8; ++i) xRs[m * XSTR + mo * 8 + i] = r[i] / den[gof(i)];
    }
  }
  __syncthreads();

  v8f acc = {0, 0, 0, 0, 0, 0, 0, 0};
#pragma unroll 1
  for (int kb = 0; kb < KSTEP2; ++kb) {
    const int kbase = kb * 4;
    const int k0 = kbase + 2 * half;
    v2f a = {0.0f, 0.0f};
    if (kbase < KGEO2) {
      int n = k0 >> 6, i = (k0 >> 3) & 7, kk = k0 & 7;
      float xv = xIn[c0 * XSTR + n * 8 + i];
      a.x = xv * xRs[c0 * XSTR + n * 8 + kk];
      a.y = xv * xRs[c0 * XSTR + n * 8 + kk + 1];
    } else if (kbase < KLEND2) {
      int t = k0 - KGEO2; int n = t >> 3, jp = t & 7;
      a.x = xIn[c0 * XSTR + n * 8 + jp];
      a.y = xIn[c0 * XSTR + n * 8 + jp + 1];
    } else if (kbase == KLEND2) {
      a.x = (half == 0) ? 1.0f : 0.0f;   // bias
    }
    v2f b = *reinterpret_cast<const v2f*>(wp2 + 2 * ((size_t)(kb * 2 + half) * 16 + c0));
    acc = wmma4(a, b, acc);
  }

  if (c0 == 7) {  // column j=7 of channel o=0 -> pseudoscalar output
    const float dp = detP[0];
#pragma unroll
    for (int r = 0; r < 8; ++r) {
      const int row = rowbase + r + 8 * half;
      const float y = acc[r] * INV_SQRT2 / dp;
      const float d = y - products[row];
      out[1 + row] = d * d;
    }
  }
}

__global__ __launch_bounds__(256) void reduce_mean(
    const float* __restrict__ loss, float* __restrict__ out0) {
  __shared__ float s[256];
  const int tid = threadIdx.x;
  float acc = 0.0f;
  for (int i = tid; i < B_PTS; i += 256) acc += loss[i];
  s[tid] = acc;
  __syncthreads();
  for (int st = 128; st > 0; st >>= 1) {
    if (tid < st) s[tid] += s[tid + st];
    __syncthreads();
  }
  if (tid == 0) out0[0] = s[0] * (1.0f / (float)B_PTS);
}

// ---------------------------------------------------------------------------
extern "C" void kernel_launch(void* const* d_in, const int* in_sizes, int n_in,
                              void* d_out, int out_size, void* d_ws, size_t ws_size,
                              hipStream_t stream) {
  (void)in_sizes; (void)n_in; (void)out_size; (void)ws_size;
  const float* points   = (const float*)d_in[0];
  const float* products = (const float*)d_in[1];
  const float* P        = (const float*)d_in[2];
  const float* detP     = (const float*)d_in[3];
  const float* cayley   = (const float*)d_in[4];
  const float* beta     = (const float*)d_in[5];
  const float* w_geo1   = (const float*)d_in[6];
  const float* wr1      = (const float*)d_in[7];
  const float* wl1      = (const float*)d_in[8];
  const float* bl1      = (const float*)d_in[9];
  const float* an1      = (const float*)d_in[10];
  const float* silu_a   = (const float*)d_in[11];
  const float* silu_b   = (const float*)d_in[12];
  const float* w_geo2   = (const float*)d_in[13];
  const float* wr2      = (const float*)d_in[14];
  const float* wl2      = (const float*)d_in[15];
  const float* bl2      = (const float*)d_in[16];
  const float* an2      = (const float*)d_in[17];

  char* ws = (char*)d_ws;
  float* wp1   = (float*)(ws);                                       // 112*256 float2
  float* wp2   = (float*)(ws + (size_t)QW1 * N1 * 8);                // 1154*16 float2
  float* x_mid = (float*)(ws + (size_t)QW1 * N1 * 8 + (size_t)QW2 * 16 * 8);  // 32768*256 f32
  float* out = (float*)d_out;

  const int wtot = QW1 * N1 + QW2 * 16;
  build_weights<<<(wtot + 255) / 256, 256, 0, stream>>>(cayley, w_geo1, wl1, bl1,
                                                        w_geo2, wl2, bl2, wp1, wp2);
  gemm_fcsgp1<<<B_PTS / 16, 32, 0, stream>>>(points, P, wr1, an1, beta, wp1, x_mid);
  silu5<<<(B_PTS * FH) / 256, 256, 0, stream>>>(x_mid, beta, silu_a, silu_b);
  gemm_fcsgp2<<<B_PTS / 16, 32, 0, stream>>>(x_mid, wr2, an2, beta, wp2, products, detP, out);
  reduce_mean<<<1, 256, 0, stream>>>(out + 1, out);
}